// EMA_33191507263821
// MI455X (gfx1250) — compile-verified
//
#include <hip/hip_runtime.h>

// EMA scan y_t = 0.7*y_{t-1} + 0.3*x_t over (8, 4096, 1024) fp32.
// Chunked scan: 16-step chunks; intra-chunk scan done as D = L*U via
// V_WMMA_F32_16X16X4_F32 (4 chained K=4 slices); inter-chunk carries via a
// 3-level spine scan in workspace.

#define ALPHA 0.3f
#define OMA   0.7f
#define B_    8
#define T_    4096
#define C_    1024
#define NCH   (T_ / 16)      // 256 chunks per sequence
#define NSUP  16             // superchunks (16 chunks each)
#define NCB   (C_ / 16)      // 64 channel blocks of 16

typedef __attribute__((ext_vector_type(2))) float v2f;
typedef __attribute__((ext_vector_type(8))) float v8f;

#define LOG2_07 (-0.5145731728297583f)   // log2(0.7)

// ---------------------------------------------------------------------------
// Kernel 1: per-chunk summary (carry-out with zero carry-in).
// co[b][j][c] = fold_{s=0..15}  acc = 0.7*acc + u_s,
//   u_s = x (global t==0) else ALPHA*x.
// Thread per (b, j, c); consecutive threads -> consecutive c (coalesced).
// ---------------------------------------------------------------------------
__global__ void __launch_bounds__(256) ema_summary(const float* __restrict__ x,
                                                   float* __restrict__ co) {
  int idx = blockIdx.x * 256 + threadIdx.x;      // [0, B_*NCH*C_)
  int c = idx % C_;
  int j = (idx / C_) % NCH;
  int b = idx / (C_ * NCH);
  const float* px = x + ((size_t)(b * T_ + j * 16) * C_ + c);
  float acc = 0.0f;
#pragma unroll
  for (int s = 0; s < 16; ++s) {
    float v = px[(size_t)s * C_];
    float u = (j == 0 && s == 0) ? v : ALPHA * v;
    acc = OMA * acc + u;
  }
  co[idx] = acc;   // co layout == idx layout: (b*NCH + j)*C_ + c
}

// ---------------------------------------------------------------------------
// Kernel 2a: superchunk summaries. S[b][g][c] = fold over 16 chunk summaries.
// ---------------------------------------------------------------------------
__global__ void __launch_bounds__(256) ema_spine_a(const float* __restrict__ co,
                                                   float* __restrict__ S) {
  int idx = blockIdx.x * 256 + threadIdx.x;      // [0, B_*NSUP*C_)
  int c = idx % C_;
  int g = (idx / C_) % NSUP;
  int b = idx / (C_ * NSUP);
  const float d16 = 0.0033232930570f;            // 0.7^16
  float q = 0.0f;
#pragma unroll
  for (int i = 0; i < 16; ++i) {
    q = d16 * q + co[(size_t)(b * NCH + g * 16 + i) * C_ + c];
  }
  S[(size_t)(b * NSUP + g) * C_ + c] = q;
}

// ---------------------------------------------------------------------------
// Kernel 2b: scan superchunk summaries in place -> carry into each superchunk.
// ---------------------------------------------------------------------------
__global__ void __launch_bounds__(256) ema_spine_b(float* __restrict__ S) {
  int idx = blockIdx.x * 256 + threadIdx.x;      // [0, B_*C_)
  int c = idx % C_;
  int b = idx / C_;
  const float d128 = 1.487810e-20f;              // 0.7^128
  const float d256 = d128 * d128;                // 0.7^256 (denormal; ~0)
  float p = 0.0f;
#pragma unroll
  for (int g = 0; g < NSUP; ++g) {
    size_t off = (size_t)(b * NSUP + g) * C_ + c;
    float t = S[off];
    S[off] = p;                                  // carry INTO superchunk g
    p = d256 * p + t;
  }
}

// ---------------------------------------------------------------------------
// Kernel 2c: expand to per-chunk carries, in place in co.
// ---------------------------------------------------------------------------
__global__ void __launch_bounds__(256) ema_spine_c(float* __restrict__ co,
                                                   const float* __restrict__ S) {
  int idx = blockIdx.x * 256 + threadIdx.x;      // [0, B_*NSUP*C_)
  int c = idx % C_;
  int g = (idx / C_) % NSUP;
  int b = idx / (C_ * NSUP);
  const float d16 = 0.0033232930570f;
  float p = S[(size_t)(b * NSUP + g) * C_ + c];  // carry into superchunk g
#pragma unroll
  for (int i = 0; i < 16; ++i) {
    size_t off = (size_t)(b * NCH + g * 16 + i) * C_ + c;
    float t = co[off];
    co[off] = p;                                 // carry INTO chunk g*16+i
    p = d16 * p + t;
  }
}

// ---------------------------------------------------------------------------
// Kernel 3: apply. One wave per 16(t) x 16(c) tile.
// Local scan D = L * U via 4x V_WMMA_F32_16X16X4_F32 (K slices of 4), then
// y[t][c] = D[t][c] + carry[c] * 0.7^(t+1); non-temporal stores keep the
// streaming y output from evicting L2-resident x.
//
// ISA layouts (CDNA5 ISA §7.12.2, f32, wave32):
//   A 16x4:  lane m (both halves) holds row m; VGPR0 = {K0 | K2}, VGPR1 = {K1 | K3}
//   B 4x16:  rows striped across lanes per VGPR: VGPR0 = {row0 | row2}, VGPR1 = {row1 | row3}
//   C/D 16x16: VGPR v: lanes 0-15 -> row v, lanes 16-31 -> row v+8; col = lane&15
//
// The L matrix is identical for every tile: build the 4 A fragments once,
// branchlessly, with v_exp_f32 (0.7^e = exp2(e*log2 0.7)) -- no table loads.
// ---------------------------------------------------------------------------
__global__ void __launch_bounds__(256) ema_apply(const float* __restrict__ x,
                                                 const float* __restrict__ carry,
                                                 float* __restrict__ y) {
  const int wave = blockIdx.x * (256 / 32) + (threadIdx.x >> 5);
  const int lane = threadIdx.x & 31;
  const int half = lane >> 4;
  const int col  = lane & 15;

  int cb = wave % NCB;
  int j  = (wave / NCB) % NCH;
  int b  = wave / (NCB * NCH);
  const int t0 = j * 16;
  const int c0 = cb * 16;
  const size_t xbase = ((size_t)(b * T_) + t0) * C_ + c0 + col;

  // ---- A fragments (wave-invariant, computed once, pure VALU) ----
  v2f Afrag[4];
#pragma unroll
  for (int kk = 0; kk < 4; ++kk) {
    const int s0 = 4 * kk + 2 * half;            // K index for .x
    const int e0 = col - s0;
    const int e1 = e0 - 1;                       // K index s0+1 for .y
    float p0 = __builtin_amdgcn_exp2f((float)e0 * LOG2_07);
    float p1 = __builtin_amdgcn_exp2f((float)e1 * LOG2_07);
    Afrag[kk].x = (e0 >= 0) ? p0 : 0.0f;
    Afrag[kk].y = (e1 >= 0) ? p1 : 0.0f;
  }

  // ---- load the 16x16 U tile rows this lane owns (8 loads, one clause) ----
  float xv[8];
#pragma unroll
  for (int kk = 0; kk < 4; ++kk) {
    const int s0 = 4 * kk + 2 * half;
    xv[2 * kk + 0] = x[xbase + (size_t)(s0 + 0) * C_];
    xv[2 * kk + 1] = x[xbase + (size_t)(s0 + 1) * C_];
  }

  v8f acc = {0.f, 0.f, 0.f, 0.f, 0.f, 0.f, 0.f, 0.f};

#if __has_builtin(__builtin_amdgcn_wmma_f32_16x16x4_f32)
#pragma unroll
  for (int kk = 0; kk < 4; ++kk) {
    const int s0 = 4 * kk + 2 * half;
    v2f bb;
    bb.x = (t0 + s0 == 0) ? xv[2 * kk] : ALPHA * xv[2 * kk];
    bb.y = ALPHA * xv[2 * kk + 1];               // t0+s0+1 >= 1 always
    acc = __builtin_amdgcn_wmma_f32_16x16x4_f32(
        false, Afrag[kk], false, bb, (short)0, acc, false, false);
  }
#else
  // VALU fallback: each lane evaluates its 8 output rows directly.
  float u[16];
#pragma unroll
  for (int s = 0; s < 16; ++s) {
    float v = x[((size_t)(b * T_) + t0 + s) * C_ + c0 + col];
    u[s] = (t0 + s == 0) ? v : ALPHA * v;
  }
#pragma unroll
  for (int v = 0; v < 8; ++v) {
    int r = v + 8 * half;
    float d = 0.0f;
#pragma unroll
    for (int s = 0; s < 16; ++s)
      if (s <= r)
        d += __builtin_amdgcn_exp2f((float)(r - s) * LOG2_07) * u[s];
    acc[v] = d;
  }
#endif

  // Carry into this chunk for this channel.
  const float cin = carry[(size_t)(b * NCH + j) * C_ + c0 + col];

  // dec = 0.7^(r+1) for r = v + 8*half, updated by *0.7 per v.
  float dec = __builtin_amdgcn_exp2f((float)(8 * half + 1) * LOG2_07);
#pragma unroll
  for (int v = 0; v < 8; ++v) {
    const int r = v + 8 * half;                  // D-matrix row for this VGPR
    const float val = acc[v] + cin * dec;
    __builtin_nontemporal_store(
        val, &y[((size_t)(b * T_) + t0 + r) * C_ + c0 + col]);
    dec *= OMA;
  }
}

// ---------------------------------------------------------------------------
extern "C" void kernel_launch(void* const* d_in, const int* in_sizes, int n_in,
                              void* d_out, int out_size, void* d_ws,
                              size_t ws_size, hipStream_t stream) {
  (void)in_sizes; (void)n_in; (void)out_size; (void)ws_size;
  const float* x = (const float*)d_in[0];
  float* y = (float*)d_out;

  float* co = (float*)d_ws;                         // B_*NCH*C_ floats (8 MiB)
  float* S  = co + (size_t)B_ * NCH * C_;           // B_*NSUP*C_ floats (512 KiB)

  // 1) chunk summaries (reads all of x; fills L2)
  ema_summary<<<(B_ * NCH * C_) / 256, 256, 0, stream>>>(x, co);
  // 2) spine scan: chunk carries (in place in co)
  ema_spine_a<<<(B_ * NSUP * C_) / 256, 256, 0, stream>>>(co, S);
  ema_spine_b<<<(B_ * C_) / 256, 256, 0, stream>>>(S);
  ema_spine_c<<<(B_ * NSUP * C_) / 256, 256, 0, stream>>>(co, S);
  // 3) apply: WMMA local scan + carry, NT stores (x re-read hits L2)
  ema_apply<<<(B_ * NCH * NCB) / (256 / 32), 256, 0, stream>>>(x, co, y);
}